// ROIHead_4054449127651
// MI455X (gfx1250) — compile-verified
//
#include <hip/hip_runtime.h>
#include <hip/hip_bf16.h>

#define CCH    512
#define HF     50
#define WF     50
#define NPROP  2000
#define DFEAT  25088     // 512*7*7
#define FHID   1024
#define NCLS   21
#define NCAND  40000     // 2000*20
#define PRENMS 1000
#define TOPK   100
#define BBOX_CLIP 4.135166556742356f  // log(1000/16)

#define BM 128
#define BN 64
#define BK 32
#define LDP 36           // padded LDS row stride (floats): conflict-free, 16B-aligned

typedef float v2f __attribute__((ext_vector_type(2)));
typedef float v8f __attribute__((ext_vector_type(8)));
typedef int   v4i __attribute__((vector_size(16)));   // pointee type of async-LDS builtins

// ---------------------------------------------------------------------------
// Async Global->LDS copy (CDNA5 GLOBAL_LOAD_ASYNC_TO_LDS_B128, ASYNCcnt),
// with a synchronous fallback if the builtin is unavailable.
// ---------------------------------------------------------------------------
#if defined(__gfx1250__) && __has_builtin(__builtin_amdgcn_global_load_async_to_lds_b128)
#define HAVE_ASYNC_LDS 1
typedef __attribute__((address_space(1))) v4i* gv4p;  // global int4*
typedef __attribute__((address_space(3))) v4i* lv4p;  // LDS int4* (32-bit ptr)
__device__ __forceinline__ void async_copy_b128(const float* g, float* l) {
    // LDS flat address low 32 bits == DS offset (aperture rule)
    __builtin_amdgcn_global_load_async_to_lds_b128(
        (gv4p)(unsigned long long)g,
        (lv4p)(unsigned int)(unsigned long long)l,
        0, 0);
}
__device__ __forceinline__ void async_wait0() {
#if __has_builtin(__builtin_amdgcn_s_wait_asynccnt)
    __builtin_amdgcn_s_wait_asynccnt(0);
#else
    asm volatile("s_wait_asynccnt 0x0" ::: "memory");
#endif
}
#else
#define HAVE_ASYNC_LDS 0
__device__ __forceinline__ void async_copy_b128(const float* g, float* l) {
    float4 v = *(const float4*)g;
    *(float4*)l = v;
}
__device__ __forceinline__ void async_wait0() {}
#endif

// ---------------------------------------------------------------------------
// 1. ROI max-pool:  feat[512,50,50] x proposals[2000,4] -> pooled[2000,25088]
// ---------------------------------------------------------------------------
__global__ __launch_bounds__(256) void roi_pool_kernel(
    const float* __restrict__ feat, const float* __restrict__ props,
    const int* __restrict__ imshape, float* __restrict__ pooled)
{
    int n = blockIdx.x;
    float imgH = (float)imshape[0];
    float scale = exp2f(rintf(log2f((float)HF / imgH)));
    float rx1 = rintf(props[n * 4 + 0] * scale);
    float ry1 = rintf(props[n * 4 + 1] * scale);
    float rx2 = rintf(props[n * 4 + 2] * scale);
    float ry2 = rintf(props[n * 4 + 3] * scale);
    float rw = fmaxf(rx2 - rx1 + 1.0f, 1.0f);
    float rh = fmaxf(ry2 - ry1 + 1.0f, 1.0f);
    for (int o = threadIdx.x; o < DFEAT; o += 256) {
        int c = o / 49;
        int bin = o - c * 49;
        int ph = bin / 7, pw = bin - ph * 7;
        int hs = (int)fminf(fmaxf(floorf((float)ph * rh / 7.0f) + ry1, 0.0f), (float)HF);
        int he = (int)fminf(fmaxf(ceilf((float)(ph + 1) * rh / 7.0f) + ry1, 0.0f), (float)HF);
        int ws = (int)fminf(fmaxf(floorf((float)pw * rw / 7.0f) + rx1, 0.0f), (float)WF);
        int we = (int)fminf(fmaxf(ceilf((float)(pw + 1) * rw / 7.0f) + rx1, 0.0f), (float)WF);
        float m = -1e30f;
        const float* fc = feat + (size_t)c * (HF * WF);
        for (int h = hs; h < he; ++h)
            for (int w = ws; w < we; ++w)
                m = fmaxf(m, fc[h * WF + w]);
        if (he <= hs || we <= ws) m = 0.0f;
        pooled[(size_t)n * DFEAT + o] = m;
    }
}

// ---------------------------------------------------------------------------
// 2. LDS-staged, double-buffered fp32 WMMA GEMM:
//    out[M,N] = A[M,K] * W[N,K]^T + bias, K % 32 == 0.
//    Block = 256 thr = 8 waves; tile BM=128 x BN=64 x BK=32.
//    Wave w -> rows [16w,16w+16) x all 64 cols: 4 x v_wmma_f32_16x16x4_f32
//    per K-step, fragments read from LDS; next K-block DMA'd asynchronously.
// ---------------------------------------------------------------------------
__device__ __forceinline__ void issue_tile(
    const float* __restrict__ A, const float* __restrict__ W,
    float* sA, float* sB, int m0, int n0, int k0,
    int M, int N, int K, int tid)
{
#pragma unroll
    for (int i = 0; i < 4; ++i) {                  // A: 128 rows * 8 chunks
        int c = tid + i * 256;
        int row = c >> 3, kc = c & 7;
        int gr = m0 + row; gr = (gr < M) ? gr : (M - 1);
        async_copy_b128(A + (size_t)gr * K + k0 + kc * 4, sA + row * LDP + kc * 4);
    }
#pragma unroll
    for (int i = 0; i < 2; ++i) {                  // B: 64 cols * 8 chunks
        int c = tid + i * 256;
        int col = c >> 3, kc = c & 7;
        int gc = n0 + col; gc = (gc < N) ? gc : (N - 1);
        async_copy_b128(W + (size_t)gc * K + k0 + kc * 4, sB + col * LDP + kc * 4);
    }
}

__global__ __launch_bounds__(256) void gemm_wmma_kernel(
    const float* __restrict__ A, const float* __restrict__ W,
    const float* __restrict__ bias, float* __restrict__ out,
    int M, int N, int K, int nBlocks, int relu)
{
    __shared__ float sA[2][BM * LDP];
    __shared__ float sB[2][BN * LDP];

    int tid  = threadIdx.x;
    int wave = tid >> 5;
    int lane = tid & 31;
    int lr = lane & 15;                   // fragment row / column
    int hi = lane >> 4;                   // K-pair select (A/B 32-bit layout)
    int mB = blockIdx.x / nBlocks;
    int nB = blockIdx.x - mB * nBlocks;
    int m0 = mB * BM;
    int n0 = nB * BN;

    v8f acc[4] = {};
    int nK = K / BK;

    issue_tile(A, W, sA[0], sB[0], m0, n0, 0, M, N, K, tid);

    int pb = 0;
    for (int kb = 0; kb < nK; ++kb) {
        async_wait0();                    // own async batch into buf pb done
        __syncthreads();                  // everyone's batch done + prev compute done
        if (kb + 1 < nK)
            issue_tile(A, W, sA[pb ^ 1], sB[pb ^ 1], m0, n0, (kb + 1) * BK,
                       M, N, K, tid);     // DMA overlaps the math below

        const float* sAw = &sA[pb][(wave * 16 + lr) * LDP + 2 * hi];
        const float* sB0 = &sB[pb][lr * LDP + 2 * hi];
#pragma unroll
        for (int kk = 0; kk < BK; kk += 4) {
            v2f a = *(const v2f*)(sAw + kk);
#pragma unroll
            for (int t = 0; t < 4; ++t) {
                v2f b = *(const v2f*)(sB0 + t * 16 * LDP + kk);
                acc[t] = __builtin_amdgcn_wmma_f32_16x16x4_f32(
                    false, a, false, b, (short)0, acc[t], false, false);
            }
        }
        pb ^= 1;
    }

#pragma unroll
    for (int t = 0; t < 4; ++t) {
        int c = n0 + 16 * t + lr;
        if (c < N) {
            float bv = bias[c];
#pragma unroll
            for (int r = 0; r < 8; ++r) {   // C/D layout: VGPR r -> M = r + 8*hi
                int row = m0 + wave * 16 + r + 8 * hi;
                if (row < M) {
                    float v = acc[t][r] + bv;
                    if (relu) v = fmaxf(v, 0.0f);
                    out[(size_t)row * N + c] = v;
                }
            }
        }
    }
}

// ---------------------------------------------------------------------------
// 3. Softmax + box decode + score/size filter -> 40000 candidates
// ---------------------------------------------------------------------------
__global__ __launch_bounds__(256) void decode_kernel(
    const float* __restrict__ cls, const float* __restrict__ bbox,
    const float* __restrict__ props, const int* __restrict__ imshape,
    float* __restrict__ candScore, float* __restrict__ candBox)
{
    int n = blockIdx.x * 256 + threadIdx.x;
    if (n >= NPROP) return;
    float imgH = (float)imshape[0], imgW = (float)imshape[1];

    const float* s = cls + (size_t)n * NCLS;
    float mx = s[0];
    for (int j = 1; j < NCLS; ++j) mx = fmaxf(mx, s[j]);
    float e[NCLS], sum = 0.0f;
    for (int j = 0; j < NCLS; ++j) { e[j] = expf(s[j] - mx); sum += e[j]; }
    float inv = 1.0f / sum;

    float px1 = props[n * 4 + 0], py1 = props[n * 4 + 1];
    float px2 = props[n * 4 + 2], py2 = props[n * 4 + 3];
    float pw = px2 - px1, phh = py2 - py1;
    float pcx = px1 + 0.5f * pw, pcy = py1 + 0.5f * phh;
    const float* d = bbox + (size_t)n * (NCLS * 4);

    for (int j = 1; j < NCLS; ++j) {        // drop background class 0
        float dx = d[j * 4 + 0], dy = d[j * 4 + 1];
        float dw = d[j * 4 + 2], dh = d[j * 4 + 3];
        float gcx = dx * pw + pcx, gcy = dy * phh + pcy;
        float gw = expf(fminf(dw, BBOX_CLIP)) * pw;
        float gh = expf(fminf(dh, BBOX_CLIP)) * phh;
        float bx1 = fminf(fmaxf(gcx - 0.5f * gw, 0.0f), imgW);
        float by1 = fminf(fmaxf(gcy - 0.5f * gh, 0.0f), imgH);
        float bx2 = fminf(fmaxf(gcx + 0.5f * gw, 0.0f), imgW);
        float by2 = fminf(fmaxf(gcy + 0.5f * gh, 0.0f), imgH);
        float sc = e[j] * inv;
        bool valid = (sc > 0.05f) && (bx2 - bx1 >= 1.0f) && (by2 - by1 >= 1.0f);
        int idx = n * 20 + (j - 1);
        candScore[idx] = valid ? sc : -1.0f;
        candBox[idx * 4 + 0] = bx1;
        candBox[idx * 4 + 1] = by1;
        candBox[idx * 4 + 2] = bx2;
        candBox[idx * 4 + 3] = by2;
    }
}

// ---------------------------------------------------------------------------
// 4. Deterministic top-1000 (score desc, ties -> lowest index, like top_k)
// ---------------------------------------------------------------------------
__global__ __launch_bounds__(1024) void select_topk_kernel(
    float* __restrict__ candScore, const float* __restrict__ candBox,
    const int* __restrict__ imshape, float* __restrict__ selScore,
    float* __restrict__ selBox, float* __restrict__ selObox,
    float* __restrict__ selLabel)
{
    __shared__ float sv[1024];
    __shared__ int   si[1024];
    int t = threadIdx.x;
    float off = (float)((imshape[0] > imshape[1]) ? imshape[0] : imshape[1]) + 1.0f;

    for (int it = 0; it < PRENMS; ++it) {
        float bv = -1e38f; int bi = 0x7fffffff;
        for (int j = t; j < NCAND; j += 1024) {
            float v = candScore[j];
            if (v > bv || (v == bv && j < bi)) { bv = v; bi = j; }
        }
        sv[t] = bv; si[t] = bi;
        __syncthreads();
        for (int srd = 512; srd > 0; srd >>= 1) {
            if (t < srd) {
                float ov = sv[t + srd]; int oi = si[t + srd];
                if (ov > sv[t] || (ov == sv[t] && oi < si[t])) { sv[t] = ov; si[t] = oi; }
            }
            __syncthreads();
        }
        if (t == 0) {
            int j = si[0];
            selScore[it] = sv[0];
            float b0 = candBox[j * 4 + 0], b1 = candBox[j * 4 + 1];
            float b2 = candBox[j * 4 + 2], b3 = candBox[j * 4 + 3];
            selBox[it * 4 + 0] = b0; selBox[it * 4 + 1] = b1;
            selBox[it * 4 + 2] = b2; selBox[it * 4 + 3] = b3;
            float lab = (float)((j % 20) + 1);
            selLabel[it] = lab;
            float o = lab * off;                 // per-class offset trick
            selObox[it * 4 + 0] = b0 + o; selObox[it * 4 + 1] = b1 + o;
            selObox[it * 4 + 2] = b2 + o; selObox[it * 4 + 3] = b3 + o;
            candScore[j] = -3.0f;                // below the -1 invalid sentinel
        }
        __syncthreads();
    }
}

// ---------------------------------------------------------------------------
// 5. Greedy NMS in LDS + final top-100 compaction into d_out
//    (candidates already score-sorted -> first 100 survivors == final top-100)
// ---------------------------------------------------------------------------
__global__ __launch_bounds__(1024) void nms_finalize_kernel(
    const float* __restrict__ selScore, const float* __restrict__ selBox,
    const float* __restrict__ selObox, const float* __restrict__ selLabel,
    float* __restrict__ outp)
{
    __shared__ float ob[PRENMS * 4];
    __shared__ float ar[PRENMS];
    __shared__ int   sup[PRENMS];
    int t = threadIdx.x;
    for (int j = t; j < PRENMS; j += 1024) {
        float x1 = selObox[j * 4 + 0], y1 = selObox[j * 4 + 1];
        float x2 = selObox[j * 4 + 2], y2 = selObox[j * 4 + 3];
        ob[j * 4 + 0] = x1; ob[j * 4 + 1] = y1;
        ob[j * 4 + 2] = x2; ob[j * 4 + 3] = y2;
        ar[j] = (x2 - x1) * (y2 - y1);
        sup[j] = (selScore[j] > 0.0f) ? 0 : 1;   // invalid rows pre-suppressed
    }
    for (int j = t; j < 4 * TOPK + 2 * TOPK; j += 1024) outp[j] = 0.0f;

    float mx1 = 0, my1 = 0, mx2 = 0, my2 = 0, ma = 0;
    if (t < PRENMS) {
        mx1 = ob[t * 4 + 0]; my1 = ob[t * 4 + 1];
        mx2 = ob[t * 4 + 2]; my2 = ob[t * 4 + 3];
        ma = ar[t];
    }
    for (int i = 0; i < PRENMS; ++i) {
        __syncthreads();
        int supi = sup[i];
        if (!supi && t > i && t < PRENMS) {
            float xx1 = fmaxf(mx1, ob[i * 4 + 0]);
            float yy1 = fmaxf(my1, ob[i * 4 + 1]);
            float xx2 = fminf(mx2, ob[i * 4 + 2]);
            float yy2 = fminf(my2, ob[i * 4 + 3]);
            float w = fmaxf(xx2 - xx1, 0.0f);
            float h = fmaxf(yy2 - yy1, 0.0f);
            float inter = w * h;
            float iou = inter / (ma + ar[i] - inter + 1e-9f);
            if (iou > 0.5f) sup[t] = 1;
        }
    }
    __syncthreads();
    if (t == 0) {
        int cnt = 0;
        for (int j = 0; j < PRENMS && cnt < TOPK; ++j) {
            if (!sup[j] && selScore[j] > 0.0f) {
                outp[cnt * 4 + 0] = selBox[j * 4 + 0];
                outp[cnt * 4 + 1] = selBox[j * 4 + 1];
                outp[cnt * 4 + 2] = selBox[j * 4 + 2];
                outp[cnt * 4 + 3] = selBox[j * 4 + 3];
                outp[4 * TOPK + cnt] = selScore[j];
                outp[5 * TOPK + cnt] = selLabel[j];
                ++cnt;
            }
        }
    }
}

// ---------------------------------------------------------------------------
extern "C" void kernel_launch(void* const* d_in, const int* in_sizes, int n_in,
                              void* d_out, int out_size, void* d_ws, size_t ws_size,
                              hipStream_t stream) {
    const float* feat    = (const float*)d_in[0];
    const float* props   = (const float*)d_in[1];
    const int*   imshape = (const int*)d_in[2];
    const float* fc6_w   = (const float*)d_in[3];
    const float* fc6_b   = (const float*)d_in[4];
    const float* fc7_w   = (const float*)d_in[5];
    const float* fc7_b   = (const float*)d_in[6];
    const float* cls_w   = (const float*)d_in[7];
    const float* cls_b   = (const float*)d_in[8];
    const float* bbox_w  = (const float*)d_in[9];
    const float* bbox_b  = (const float*)d_in[10];
    float* outp = (float*)d_out;

    // workspace layout (float offsets)
    float* wsf = (float*)d_ws;
    float* pooled    = wsf;                                   // 2000*25088
    float* x1        = pooled + (size_t)NPROP * DFEAT;        // 2000*1024
    float* x2        = x1 + (size_t)NPROP * FHID;             // 2000*1024
    float* clsOut    = x2 + (size_t)NPROP * FHID;             // 2000*21
    float* bboxOut   = clsOut + (size_t)NPROP * NCLS;         // 2000*84
    float* candScore = bboxOut + (size_t)NPROP * NCLS * 4;    // 40000
    float* candBox   = candScore + NCAND;                     // 40000*4
    float* selScore  = candBox + (size_t)NCAND * 4;           // 1000
    float* selBox    = selScore + PRENMS;                     // 1000*4
    float* selObox   = selBox + PRENMS * 4;                   // 1000*4
    float* selLabel  = selObox + PRENMS * 4;                  // 1000

    int mBlocks = (NPROP + BM - 1) / BM;                      // 16

    // 1. ROI max-pool
    roi_pool_kernel<<<NPROP, 256, 0, stream>>>(feat, props, imshape, pooled);

    // 2. FC6 (relu): [2000,25088] x [25088,1024]
    {
        int nBlocks = (FHID + BN - 1) / BN;                   // 16
        gemm_wmma_kernel<<<mBlocks * nBlocks, 256, 0, stream>>>(
            pooled, fc6_w, fc6_b, x1, NPROP, FHID, DFEAT, nBlocks, 1);
    }
    // 3. FC7 (relu): [2000,1024] x [1024,1024]
    {
        int nBlocks = (FHID + BN - 1) / BN;
        gemm_wmma_kernel<<<mBlocks * nBlocks, 256, 0, stream>>>(
            x1, fc7_w, fc7_b, x2, NPROP, FHID, FHID, nBlocks, 1);
    }
    // 4. cls head: [2000,1024] x [1024,21]
    {
        int nBlocks = (NCLS + BN - 1) / BN;                   // 1
        gemm_wmma_kernel<<<mBlocks * nBlocks, 256, 0, stream>>>(
            x2, cls_w, cls_b, clsOut, NPROP, NCLS, FHID, nBlocks, 0);
    }
    // 5. bbox head: [2000,1024] x [1024,84]
    {
        int nBlocks = (NCLS * 4 + BN - 1) / BN;               // 2
        gemm_wmma_kernel<<<mBlocks * nBlocks, 256, 0, stream>>>(
            x2, bbox_w, bbox_b, bboxOut, NPROP, NCLS * 4, FHID, nBlocks, 0);
    }
    // 6. softmax + box decode + filter
    decode_kernel<<<(NPROP + 255) / 256, 256, 0, stream>>>(
        clsOut, bboxOut, props, imshape, candScore, candBox);
    // 7. top-1000 selection
    select_topk_kernel<<<1, 1024, 0, stream>>>(
        candScore, candBox, imshape, selScore, selBox, selObox, selLabel);
    // 8. greedy NMS + top-100 compaction into d_out
    nms_finalize_kernel<<<1, 1024, 0, stream>>>(
        selScore, selBox, selObox, selLabel, outp);
}